// Block_25039659335700
// MI455X (gfx1250) — compile-verified
//
#include <hip/hip_runtime.h>
#include <hip/hip_bf16.h>
#include <math.h>
#include <stdint.h>

#define Bb 16
#define Tt 512
#define Cc 1024
#define Hh 16
#define HDd 64

typedef float v2f __attribute__((ext_vector_type(2)));
typedef float v8f __attribute__((ext_vector_type(8)));
typedef uint32_t u32x4 __attribute__((ext_vector_type(4)));
typedef int i32x4 __attribute__((ext_vector_type(4)));
typedef int i32x8 __attribute__((ext_vector_type(8)));

#if __has_builtin(__builtin_amdgcn_tensor_load_to_lds) && \
    __has_builtin(__builtin_amdgcn_s_wait_tensorcnt)
#define USE_TDM 1
#else
#define USE_TDM 0
#endif

// D = A(16x4,f32) * B(4x16,f32) + C(16x16,f32)  -- full fp32 WMMA on CDNA5
static __device__ __forceinline__ v8f wmma4(v2f a, v2f b, v8f c) {
  return __builtin_amdgcn_wmma_f32_16x16x4_f32(false, a, false, b, (short)0, c,
                                               false, false);
}

static __device__ __forceinline__ float gelu_f(float x) {
  const float k = 0.7978845608028654f; // sqrt(2/pi)
  float x3 = x * x * x;
  return 0.5f * x * (1.0f + tanhf(k * (x + 0.044715f * x3)));
}

#if USE_TDM
// Issue one TDM 2D tile load: global (tile_y rows x tile_x f32, row stride
// stride_elts) -> LDS at lds_byte_addr, with LDS row padding
// (pad codes per D# spec: interval 2<<i dwords, amount a+1 dwords).
// Toolchain here is the 6-arg builtin flavor (clang-23 / therock-10.0):
//   (u32x4 g0, i32x8 g1, i32x4 g2, i32x4 g3, i32x8 extra, i32 cpol)
static __device__ __forceinline__ void tdm_load_2d(uint32_t lds_byte_addr,
                                                   const float* gaddr,
                                                   uint32_t tile_x,
                                                   uint32_t tile_y,
                                                   uint32_t stride_elts,
                                                   uint32_t pad_interval_code,
                                                   uint32_t pad_amount_code) {
  uint64_t ga = (uint64_t)(uintptr_t)gaddr;
  u32x4 g0;
  g0[0] = 1u;                                  // count=1 (valid user D#)
  g0[1] = lds_byte_addr;                       // lds_addr
  g0[2] = (uint32_t)(ga & 0xFFFFFFFFu);        // global_addr[31:0]
  g0[3] = (uint32_t)((ga >> 32) & 0x1FFFFFFu)  // global_addr[56:32]
          | (2u << 30);                        // type=2 ("image")
  i32x8 g1;
  uint32_t d0 = (2u << 16)                     // data_size = 4 bytes
                | (1u << 20)                   // pad_enable
                | (pad_interval_code << 22) | (pad_amount_code << 25);
  g1[0] = (int)d0;
  g1[1] = (int)((tile_x & 0xFFFFu) << 16);                    // tensor_dim0 lo
  g1[2] = (int)((tile_x >> 16) | ((tile_y & 0xFFFFu) << 16)); // dim0 hi, dim1 lo
  g1[3] = (int)((tile_y >> 16) | (tile_x << 16));             // dim1 hi, tile_dim0
  g1[4] = (int)(tile_y & 0xFFFFu);                            // tile_dim1 (dim2=0)
  g1[5] = (int)stride_elts;                                   // dim0_stride lo
  g1[6] = 0;
  g1[7] = 0;
  i32x4 z4 = {0, 0, 0, 0};
  i32x8 z8 = {0, 0, 0, 0, 0, 0, 0, 0};
  __builtin_amdgcn_tensor_load_to_lds(g0, g1, z4, z4, z8, 0);
}
#define LDS_OFF(p) ((uint32_t)(uintptr_t)(p))
#endif

// ---------------- LayerNorm: one 256-thread block per row (C=1024) ----------
__global__ __launch_bounds__(256) void ln_kernel(const float* __restrict__ src,
                                                 const float* __restrict__ g,
                                                 const float* __restrict__ bt,
                                                 float* __restrict__ dst) {
  const int row = blockIdx.x;
  const int tid = threadIdx.x;
  __shared__ float red[256];

  const float4* s4 = (const float4*)(src + (size_t)row * Cc);
  float4 v = s4[tid];

  red[tid] = v.x + v.y + v.z + v.w;
  __syncthreads();
  for (int s = 128; s > 0; s >>= 1) {
    if (tid < s) red[tid] += red[tid + s];
    __syncthreads();
  }
  const float mu = red[0] * (1.0f / Cc);
  __syncthreads();

  float dx = v.x - mu, dy = v.y - mu, dz = v.z - mu, dw = v.w - mu;
  red[tid] = dx * dx + dy * dy + dz * dz + dw * dw;
  __syncthreads();
  for (int s = 128; s > 0; s >>= 1) {
    if (tid < s) red[tid] += red[tid + s];
    __syncthreads();
  }
  const float rstd = rsqrtf(red[0] * (1.0f / Cc) + 1e-5f);

  float4 g4 = ((const float4*)g)[tid];
  float4 b4 = ((const float4*)bt)[tid];
  float4 o;
  o.x = dx * rstd * g4.x + b4.x;
  o.y = dy * rstd * g4.y + b4.y;
  o.z = dz * rstd * g4.z + b4.z;
  o.w = dw * rstd * g4.w + b4.w;
  ((float4*)(dst + (size_t)row * Cc))[tid] = o;
}

// ---------------- Tiled WMMA GEMM: D = act(A @ B + bias [+ res]) ------------
// 128x128x32 tiles, 256 threads (8 waves, 4(M)x2(N) wave grid), TDM double-
// buffered tile staging: wave0 DMAs tile kt+1 while all waves compute kt.
#define BM 128
#define BN 128
#define BKk 32
#define LDA 36   // = 32 + 4  (TDM pad codes 4/3) : conflict-free A-frag reads
#define LDB 132  // = 128 + 4 (TDM pad codes 6/3)

__global__ __launch_bounds__(256) void gemm_kernel(
    const float* __restrict__ A, const float* __restrict__ Bw,
    const float* __restrict__ bias, const float* __restrict__ res,
    float* __restrict__ D, int M, int N, int K, int do_gelu) {
  __shared__ float As[2][BM * LDA];
  __shared__ float Bs[2][BKk * LDB];

  const int tid = threadIdx.x;
  const int lane = tid & 31;
  const int wid = tid >> 5;
  const int hi = lane >> 4;
  const int ln16 = lane & 15;
  const int waveM = wid >> 1; // 0..3, 32 rows each
  const int waveN = wid & 1;  // 0..1, 64 cols each
  const int rowBase = blockIdx.y * BM;
  const int colBase = blockIdx.x * BN;

  v8f acc[2][4];
#pragma unroll
  for (int i = 0; i < 2; ++i)
#pragma unroll
    for (int j = 0; j < 4; ++j)
      acc[i][j] = (v8f){0.f, 0.f, 0.f, 0.f, 0.f, 0.f, 0.f, 0.f};

  const int ktiles = K / BKk;

#if USE_TDM
  if (tid < 32) { // wave 0 drives the Tensor Data Mover
    tdm_load_2d(LDS_OFF(&As[0][0]), A + (size_t)rowBase * K, BKk, BM, K, 4, 3);
    tdm_load_2d(LDS_OFF(&Bs[0][0]), Bw + colBase, BN, BKk, N, 6, 3);
  }
#endif

  for (int kt = 0; kt < ktiles; ++kt) {
    const int cur = kt & 1;
#if USE_TDM
    if (tid < 32) {
      if (kt + 1 < ktiles) {
        tdm_load_2d(LDS_OFF(&As[cur ^ 1][0]),
                    A + (size_t)rowBase * K + (size_t)(kt + 1) * BKk, BKk, BM,
                    K, 4, 3);
        tdm_load_2d(LDS_OFF(&Bs[cur ^ 1][0]),
                    Bw + (size_t)(kt + 1) * BKk * N + colBase, BN, BKk, N, 6,
                    3);
        __builtin_amdgcn_s_wait_tensorcnt(2); // tile kt fully in LDS
      } else {
        __builtin_amdgcn_s_wait_tensorcnt(0);
      }
    }
#else
#pragma unroll
    for (int i = 0; i < 4; ++i) {
      int slot = tid + i * 256;
      int r = slot >> 3, c = (slot & 7) << 2;
      float4 t = *(const float4*)(A + (size_t)(rowBase + r) * K + kt * BKk + c);
      As[cur][r * LDA + c + 0] = t.x;
      As[cur][r * LDA + c + 1] = t.y;
      As[cur][r * LDA + c + 2] = t.z;
      As[cur][r * LDA + c + 3] = t.w;
    }
#pragma unroll
    for (int i = 0; i < 4; ++i) {
      int slot = tid + i * 256;
      int r = slot >> 5, c = (slot & 31) << 2;
      float4 t = *(const float4*)(Bw + (size_t)(kt * BKk + r) * N + colBase + c);
      *(float4*)(&Bs[cur][r * LDB + c]) = t;
    }
#endif
    __syncthreads(); // tile kt visible to every wave

    const float* Asc = &As[cur][0];
    const float* Bsc = &Bs[cur][0];
#pragma unroll
    for (int k0 = 0; k0 < BKk; k0 += 4) {
      v2f a[2], bfr[4];
#pragma unroll
      for (int tm = 0; tm < 2; ++tm) {
        int r = waveM * 32 + tm * 16 + ln16;
        a[tm] = *(const v2f*)(&Asc[r * LDA + k0 + 2 * hi]);
      }
#pragma unroll
      for (int tn = 0; tn < 4; ++tn) {
        int c = waveN * 64 + tn * 16 + ln16;
        bfr[tn].x = Bsc[(k0 + 2 * hi) * LDB + c];
        bfr[tn].y = Bsc[(k0 + 2 * hi + 1) * LDB + c];
      }
#pragma unroll
      for (int tm = 0; tm < 2; ++tm)
#pragma unroll
        for (int tn = 0; tn < 4; ++tn)
          acc[tm][tn] = wmma4(a[tm], bfr[tn], acc[tm][tn]);
    }
    __syncthreads(); // buffer may be overwritten by next DMA
  }

  // epilogue: C/D layout -> reg r holds rows (r + 8*hi), col = ln16
#pragma unroll
  for (int tm = 0; tm < 2; ++tm)
#pragma unroll
    for (int tn = 0; tn < 4; ++tn) {
      int col = colBase + waveN * 64 + tn * 16 + ln16;
      float bv = bias[col];
#pragma unroll
      for (int r = 0; r < 8; ++r) {
        int row = rowBase + waveM * 32 + tm * 16 + r + 8 * hi;
        float v = acc[tm][tn][r] + bv;
        if (res) v += res[(size_t)row * N + col];
        if (do_gelu) v = gelu_f(v);
        D[(size_t)row * N + col] = v;
      }
    }
}

// ---------------- Flash attention (online softmax, WMMA S and P@V) ----------
// grid = B*H*(T/128); block = 256 (8 waves); each wave owns 16 query rows.
// K/V 16-key chunks are TDM double-buffered like the GEMM tiles.
__global__ __launch_bounds__(256) void attn_kernel(const float* __restrict__ qkv,
                                                   const int* __restrict__ amask,
                                                   float* __restrict__ y) {
  __shared__ float Ksh[2][16 * 68]; // 64 + 4 pad (TDM codes 5/3)
  __shared__ float Vsh[2][16 * 68];
  __shared__ float Pl[8 * 16 * 17]; // per-wave P transpose scratch

  const int tid = threadIdx.x;
  const int lane = tid & 31, wid = tid >> 5;
  const int hi = lane >> 4, ln16 = lane & 15;
  const int bx = blockIdx.x;
  const int bb = bx >> 6;
  const int hh = (bx >> 2) & 15;
  const int qblk = bx & 3;
  const int qbase = qblk * 128 + wid * 16;

  // preload this wave's Q rows as A-fragments (16 k-steps of 4)
  v2f qa[16];
  {
    const float* qrow =
        qkv + ((size_t)(bb * Tt + qbase + ln16)) * (3 * Cc) + hh * HDd;
#pragma unroll
    for (int s = 0; s < 16; ++s) qa[s] = *(const v2f*)(qrow + 4 * s + 2 * hi);
  }

  v8f o[4];
#pragma unroll
  for (int t = 0; t < 4; ++t)
    o[t] = (v8f){0.f, 0.f, 0.f, 0.f, 0.f, 0.f, 0.f, 0.f};
  float rmax[8], rsum[8];
#pragma unroll
  for (int r = 0; r < 8; ++r) { rmax[r] = -3e38f; rsum[r] = 0.f; }

  const float* kbase = qkv + (size_t)bb * Tt * (3 * Cc) + Cc + hh * HDd;
  const int nch = Tt / 16;

#if USE_TDM
  if (tid < 32) {
    tdm_load_2d(LDS_OFF(&Ksh[0][0]), kbase, HDd, 16, 3 * Cc, 5, 3);
    tdm_load_2d(LDS_OFF(&Vsh[0][0]), kbase + Cc, HDd, 16, 3 * Cc, 5, 3);
  }
#endif

  for (int j = 0; j < nch; ++j) {
    const int cur = j & 1;
#if USE_TDM
    if (tid < 32) {
      if (j + 1 < nch) {
        const float* nk = kbase + (size_t)(j + 1) * 16 * (3 * Cc);
        tdm_load_2d(LDS_OFF(&Ksh[cur ^ 1][0]), nk, HDd, 16, 3 * Cc, 5, 3);
        tdm_load_2d(LDS_OFF(&Vsh[cur ^ 1][0]), nk + Cc, HDd, 16, 3 * Cc, 5, 3);
        __builtin_amdgcn_s_wait_tensorcnt(2);
      } else {
        __builtin_amdgcn_s_wait_tensorcnt(0);
      }
    }
#else
    {
      int key = tid >> 4;
      int d = (tid & 15) << 2;
      const float* kp = kbase + (size_t)(j * 16 + key) * (3 * Cc) + d;
      float4 k4 = *(const float4*)kp;
      float4 v4 = *(const float4*)(kp + Cc);
      *(float4*)(&Ksh[cur][key * 68 + d]) = k4;
      *(float4*)(&Vsh[cur][key * 68 + d]) = v4;
    }
#endif
    __syncthreads();

    // S(16x16) = Q(16x64) @ K^T : B-frag b[kdim][key] = Ksh[key][kdim]
    v8f s_acc = (v8f){0.f, 0.f, 0.f, 0.f, 0.f, 0.f, 0.f, 0.f};
#pragma unroll
    for (int s = 0; s < 16; ++s) {
      v2f bfr = *(const v2f*)(&Ksh[cur][ln16 * 68 + 4 * s + 2 * hi]);
      s_acc = wmma4(qa[s], bfr, s_acc);
    }

    const int mk = amask[bb * Tt + j * 16 + ln16];
    float alpha[8];
#pragma unroll
    for (int r = 0; r < 8; ++r) {
      float sv = mk ? s_acc[r] * 0.125f : -3e38f; // 1/sqrt(64)
      float m = sv;
      for (int off = 1; off < 16; off <<= 1)
        m = fmaxf(m, __shfl_xor(m, off, 16));
      float mn = fmaxf(rmax[r], m);
      float al = expf(rmax[r] - mn);
      float p = expf(sv - mn);
      float ps = p;
      for (int off = 1; off < 16; off <<= 1) ps += __shfl_xor(ps, off, 16);
      rsum[r] = rsum[r] * al + ps;
      rmax[r] = mn;
      alpha[r] = al;
      Pl[wid * 272 + (r + 8 * hi) * 17 + ln16] = p; // C-layout -> LDS
    }
#pragma unroll
    for (int tl = 0; tl < 4; ++tl)
#pragma unroll
      for (int r = 0; r < 8; ++r) o[tl][r] = o[tl][r] * alpha[r];
    __syncthreads(); // P visible for A-layout reads

    // O(16x64) += P(16x16) @ V(16x64)
#pragma unroll
    for (int s4 = 0; s4 < 4; ++s4) {
      int k0 = 4 * s4;
      v2f a;
      a.x = Pl[wid * 272 + ln16 * 17 + k0 + 2 * hi];
      a.y = Pl[wid * 272 + ln16 * 17 + k0 + 2 * hi + 1];
#pragma unroll
      for (int tl = 0; tl < 4; ++tl) {
        v2f bfr;
        bfr.x = Vsh[cur][(k0 + 2 * hi) * 68 + tl * 16 + ln16];
        bfr.y = Vsh[cur][(k0 + 2 * hi + 1) * 68 + tl * 16 + ln16];
        o[tl] = wmma4(a, bfr, o[tl]);
      }
    }
    __syncthreads(); // done with this K/V buffer before DMA overwrites it
  }

  // normalize and write y[b][t][h*64 + d]
#pragma unroll
  for (int r = 0; r < 8; ++r) {
    float inv = 1.0f / rsum[r];
    int t = qbase + r + 8 * hi;
    float* yrow = y + ((size_t)(bb * Tt + t)) * Cc + hh * HDd;
#pragma unroll
    for (int tl = 0; tl < 4; ++tl) yrow[tl * 16 + ln16] = o[tl][r] * inv;
  }
}

// ---------------- host ------------------------------------------------------
extern "C" void kernel_launch(void* const* d_in, const int* in_sizes, int n_in,
                              void* d_out, int out_size, void* d_ws,
                              size_t ws_size, hipStream_t stream) {
  const float* x = (const float*)d_in[0];
  const int* amask = (const int*)d_in[1];
  const float* ln1g = (const float*)d_in[2];
  const float* ln1b = (const float*)d_in[3];
  const float* wattn = (const float*)d_in[4];
  const float* battn = (const float*)d_in[5];
  const float* wo = (const float*)d_in[6];
  const float* bo = (const float*)d_in[7];
  const float* ln2g = (const float*)d_in[8];
  const float* ln2b = (const float*)d_in[9];
  const float* wfc = (const float*)d_in[10];
  const float* bfc = (const float*)d_in[11];
  const float* wfc2 = (const float*)d_in[12];
  const float* bfc2 = (const float*)d_in[13];
  float* out = (float*)d_out;

  const size_t BT = (size_t)Bb * Tt; // 8192
  float* bufA = (float*)d_ws;        // BT*C   : ln1 out, later ln2 out
  float* bufB = bufA + BT * Cc;      // BT*4C  : qkv (3C), later MLP act (4C)
  float* bufC = bufB + BT * 4 * Cc;  // BT*C   : attention output y
  float* bufD = bufC + BT * Cc;      // BT*C   : x1 = x + proj

  dim3 blk(256);

  ln_kernel<<<dim3((unsigned)BT), blk, 0, stream>>>(x, ln1g, ln1b, bufA);
  gemm_kernel<<<dim3(3 * Cc / BN, (unsigned)(BT / BM)), blk, 0, stream>>>(
      bufA, wattn, battn, nullptr, bufB, (int)BT, 3 * Cc, Cc, 0);
  attn_kernel<<<dim3(Bb * Hh * (Tt / 128)), blk, 0, stream>>>(bufB, amask,
                                                              bufC);
  gemm_kernel<<<dim3(Cc / BN, (unsigned)(BT / BM)), blk, 0, stream>>>(
      bufC, wo, bo, x, bufD, (int)BT, Cc, Cc, 0);
  ln_kernel<<<dim3((unsigned)BT), blk, 0, stream>>>(bufD, ln2g, ln2b, bufA);
  gemm_kernel<<<dim3(4 * Cc / BN, (unsigned)(BT / BM)), blk, 0, stream>>>(
      bufA, wfc, bfc, nullptr, bufB, (int)BT, 4 * Cc, Cc, 1);
  gemm_kernel<<<dim3(Cc / BN, (unsigned)(BT / BM)), blk, 0, stream>>>(
      bufB, wfc2, bfc2, bufD, out, (int)BT, Cc, 4 * Cc, 0);
}